// Attention_54623394071047
// MI455X (gfx1250) — compile-verified
//
#include <hip/hip_runtime.h>
#include <math.h>

typedef __attribute__((ext_vector_type(2))) float v2f;
typedef __attribute__((ext_vector_type(8))) float v8f;

constexpr int N       = 512;
constexpr int B_DOC   = 32;
constexpr int S       = 40;
constexpr int T_K     = 64;
constexpr int B_SENT  = B_DOC * S;     // 1280
constexpr int MAXDOC  = S * T_K;       // 2560

// ---------------------------------------------------------------------------
// Kernel 1: both decoder-feature GEMMs (Y = X @ W + b) with f32 WMMA.
// One wave (32 threads) per 16x16 output tile; K-loop of 128 x V_WMMA_F32_16X16X4_F32.
// Tiles: 2 problems * (32/16) M-tiles * (512/16) N-tiles = 128 blocks of 32 thr.
// A layout (16x4 f32): lane%16 = M, half-wave selects K pair {0,1}/{2,3}.
// B layout (4x16 f32): lane%16 = N, half-wave selects K pair.
// C/D layout: VGPR r, lane l -> M = r + 8*(l/16), N = l%16.
// ---------------------------------------------------------------------------
__global__ void gemm_bias_wmma(const float* __restrict__ X0, const float* __restrict__ W0,
                               const float* __restrict__ b0,
                               const float* __restrict__ X1, const float* __restrict__ W1,
                               const float* __restrict__ b1,
                               float* __restrict__ Y0, float* __restrict__ Y1) {
  int tile = blockIdx.x;            // 0..127
  int p    = tile >> 6;             // which problem
  tile    &= 63;
  int mt   = tile >> 5;             // 0..1
  int nt   = tile & 31;             // 0..31
  const float* X    = p ? X1 : X0;
  const float* W    = p ? W1 : W0;
  const float* bias = p ? b1 : b0;
  float*       Y    = p ? Y1 : Y0;

  int lane = threadIdx.x;           // 0..31 (wave32, EXEC all ones)
  int half = lane >> 4;
  int l16  = lane & 15;

  int m  = mt * 16 + l16;           // row of A this lane carries
  int nb = nt * 16 + l16;           // col of B this lane carries

  v8f c = {};
  for (int kb = 0; kb < N / 4; ++kb) {
    int k = kb * 4 + half * 2;
    v2f a, b;
    a.x = X[m * N + k];
    a.y = X[m * N + k + 1];
    b.x = W[(k    ) * N + nb];
    b.y = W[(k + 1) * N + nb];
    c = __builtin_amdgcn_wmma_f32_16x16x4_f32(false, a, false, b, (short)0, c,
                                              false, false);
  }
  float bv = bias[nb];
#pragma unroll
  for (int r = 0; r < 8; ++r) {
    int row = mt * 16 + r + half * 8;
    Y[row * N + nb] = c[r] + bv;
  }
}

// ---------------------------------------------------------------------------
// Kernel 2: additive-attention scores, one wave per row (row length N=512):
//   out[row] = sum_n tanh(F[row,n] + D[row/div, n]) * v[n]
// Coalesced float4 streaming; this is the 168 MB encoder_feature pass.
// ---------------------------------------------------------------------------
__global__ void add_attn_scores(const float* __restrict__ F, const float* __restrict__ D,
                                const float* __restrict__ v, float* __restrict__ out,
                                int nrows, int div) {
  int wave = (int)((blockIdx.x * blockDim.x + threadIdx.x) >> 5);
  int lane = threadIdx.x & 31;
  if (wave >= nrows) return;
  const float4* f4 = (const float4*)(F + (size_t)wave * N);
  const float4* d4 = (const float4*)(D + (size_t)(wave / div) * N);
  const float4* v4 = (const float4*)v;
  float acc = 0.f;
#pragma unroll
  for (int i = 0; i < 4; ++i) {
    int idx = lane + 32 * i;
    float4 f = f4[idx], d = d4[idx], w = v4[idx];
    acc += tanhf(f.x + d.x) * w.x;
    acc += tanhf(f.y + d.y) * w.y;
    acc += tanhf(f.z + d.z) * w.z;
    acc += tanhf(f.w + d.w) * w.w;
  }
  for (int off = 16; off; off >>= 1) acc += __shfl_xor(acc, off, 32);
  if (lane == 0) out[wave] = acc;
}

// ---------------------------------------------------------------------------
// Kernel 3: per-sentence masked softmax over T_K=64 (one wave per sentence).
// softmax->mask->renorm collapses to masked softmax (partition fn cancels).
// ---------------------------------------------------------------------------
__global__ void word_softmax(const float* __restrict__ scores, const int* __restrict__ lens,
                             float* __restrict__ w_attn) {
  int sent = (int)((blockIdx.x * blockDim.x + threadIdx.x) >> 5);
  int lane = threadIdx.x & 31;
  if (sent >= B_SENT) return;
  int len = lens[sent];
  const float* s = scores + sent * T_K;
  float s0 = s[lane], s1 = s[lane + 32];
  bool m0 = lane < len, m1 = (lane + 32) < len;
  float x0 = m0 ? s0 : -INFINITY;
  float x1 = m1 ? s1 : -INFINITY;
  float mx = fmaxf(x0, x1);
  for (int off = 16; off; off >>= 1) mx = fmaxf(mx, __shfl_xor(mx, off, 32));
  float e0 = m0 ? expf(s0 - mx) : 0.f;
  float e1 = m1 ? expf(s1 - mx) : 0.f;
  float sum = e0 + e1;
  for (int off = 16; off; off >>= 1) sum += __shfl_xor(sum, off, 32);
  float inv = 1.f / sum;
  float* o = w_attn + sent * T_K;
  o[lane]      = e0 * inv;
  o[lane + 32] = e1 * inv;
}

// ---------------------------------------------------------------------------
// Kernel 4: per-doc sentence softmax over S=40 (one wave per doc), using the
// provided float mask (matches reference: full-range max, exp*mask / sum).
// ---------------------------------------------------------------------------
__global__ void sent_softmax(const float* __restrict__ scores, const float* __restrict__ mask,
                             float* __restrict__ out) {
  int doc  = (int)((blockIdx.x * blockDim.x + threadIdx.x) >> 5);
  int lane = threadIdx.x & 31;
  if (doc >= B_DOC) return;
  bool has1 = (lane + 32) < S;           // lanes 0..7 carry a second element
  float x0 = scores[doc * S + lane];
  float m0 = mask[doc * S + lane];
  float x1 = has1 ? scores[doc * S + lane + 32] : -INFINITY;
  float m1 = has1 ? mask[doc * S + lane + 32] : 0.f;
  float mx = fmaxf(x0, x1);
  for (int off = 16; off; off >>= 1) mx = fmaxf(mx, __shfl_xor(mx, off, 32));
  float e0 = expf(x0 - mx) * m0;
  float e1 = has1 ? expf(x1 - mx) * m1 : 0.f;
  float sum = e0 + e1;
  for (int off = 16; off; off >>= 1) sum += __shfl_xor(sum, off, 32);
  float inv = 1.f / sum;
  out[doc * S + lane] = e0 * inv;
  if (has1) out[doc * S + lane + 32] = e1 * inv;
}

// ---------------------------------------------------------------------------
// Kernel 5: per-doc exclusive cumsum of seq_lens2 -> word offset of each
// sentence inside its doc stream. One thread per doc (trivial size).
// ---------------------------------------------------------------------------
__global__ void doc_offsets(const int* __restrict__ lens, int* __restrict__ off) {
  int doc = blockIdx.x * blockDim.x + threadIdx.x;
  if (doc >= B_DOC) return;
  int acc = 0;
  for (int s = 0; s < S; ++s) { off[doc * S + s] = acc; acc += lens[doc * S + s]; }
}

__global__ void fill_zero(float* __restrict__ p, int n) {
  int i = blockIdx.x * blockDim.x + threadIdx.x;
  if (i < n) p[i] = 0.f;
}

// ---------------------------------------------------------------------------
// Kernel 6: scatter mult_attn = sent_attn[doc,local] * w_attn into the
// compacted per-doc stream (unique target slots; tail stays zero).
// ---------------------------------------------------------------------------
__global__ void scatter_attn(const float* __restrict__ w_attn, const float* __restrict__ sattn,
                             const int* __restrict__ lens, const int* __restrict__ off,
                             float* __restrict__ out_attn) {
  int idx = blockIdx.x * blockDim.x + threadIdx.x;   // sent*T_K + t
  if (idx >= B_SENT * T_K) return;
  int sent = idx >> 6, t = idx & 63;
  if (t >= lens[sent]) return;
  int doc = sent / S, local = sent - doc * S;
  out_attn[doc * MAXDOC + off[sent] + t] = sattn[doc * S + local] * w_attn[idx];
}

// ---------------------------------------------------------------------------
// Kernel 7: c_t partials: one block per sentence streams its 64x512 slab of
// encoder_outputs (the second 168 MB pass); weights staged in LDS.
// partial[sent,n] = sent_coef * sum_{t<len} w_attn[sent,t]*enc[sent,t,n]
// ---------------------------------------------------------------------------
__global__ void ct_partial(const float* __restrict__ enc, const float* __restrict__ w_attn,
                           const float* __restrict__ sattn, const int* __restrict__ lens,
                           float* __restrict__ partial) {
  __shared__ float sw[T_K];
  int sent = blockIdx.x;
  int tid  = threadIdx.x;                  // 0..255
  if (tid < T_K) sw[tid] = w_attn[sent * T_K + tid];
  __syncthreads();
  int len = lens[sent];
  int doc = sent / S, local = sent - doc * S;
  float coef = sattn[doc * S + local];
  const float* e = enc + (size_t)sent * T_K * N;
  float a0 = 0.f, a1 = 0.f;
  for (int t = 0; t < len; ++t) {
    float wt = sw[t];
    a0 += wt * e[(size_t)t * N + tid];
    a1 += wt * e[(size_t)t * N + tid + 256];
  }
  partial[(size_t)sent * N + tid]       = a0 * coef;
  partial[(size_t)sent * N + tid + 256] = a1 * coef;
}

// Kernel 8: reduce 40 sentence partials per doc -> c_t [32,512]
__global__ void ct_reduce(const float* __restrict__ partial, float* __restrict__ c_t) {
  int idx = blockIdx.x * blockDim.x + threadIdx.x;   // doc*N + n
  if (idx >= B_DOC * N) return;
  int doc = idx >> 9, n = idx & (N - 1);
  float acc = 0.f;
  for (int s = 0; s < S; ++s) acc += partial[(size_t)(doc * S + s) * N + n];
  c_t[idx] = acc;
}

// ---------------------------------------------------------------------------
extern "C" void kernel_launch(void* const* d_in, const int* in_sizes, int n_in,
                              void* d_out, int out_size, void* d_ws, size_t ws_size,
                              hipStream_t stream) {
  (void)in_sizes; (void)n_in; (void)out_size; (void)ws_size;
  const float* s_t_hat        = (const float*)d_in[0];
  const float* enc_outputs    = (const float*)d_in[1];   // [1280,64,512]
  const float* enc_feature    = (const float*)d_in[2];   // [81920,512]
  const int*   seq_lens2      = (const int*)  d_in[3];   // [1280]
  const float* sent_s_t_hat   = (const float*)d_in[4];
  const float* sent_enc_feat  = (const float*)d_in[6];   // [1280,512]
  const float* sent_pad_mask  = (const float*)d_in[7];   // [32,40]
  const float* coverage       = (const float*)d_in[10];  // [1280,64]
  const float* Wd  = (const float*)d_in[11];
  const float* bd  = (const float*)d_in[12];
  const float* v_w = (const float*)d_in[13];
  const float* Wsd = (const float*)d_in[14];
  const float* bsd = (const float*)d_in[15];
  const float* sv_w= (const float*)d_in[16];

  // d_out layout: [c_t 16384][attn_dist_sw 81920][coverage 81920][sent_attn 1280]
  float* out_ct    = (float*)d_out;
  float* out_attn  = out_ct + B_DOC * N;                 // +16384
  float* out_cov   = out_attn + B_DOC * MAXDOC;          // +98304
  float* out_sattn = out_cov + B_SENT * T_K;             // +180224

  // workspace carve-up (floats)
  float* ws         = (float*)d_ws;
  float* dec_fea    = ws;                                // 16384
  float* sdec_fea   = dec_fea  + B_DOC * N;              // 16384
  float* scores     = sdec_fea + B_DOC * N;              // 81920
  float* w_attn     = scores   + B_SENT * T_K;           // 81920
  float* s_scores   = w_attn   + B_SENT * T_K;           // 1280
  int*   sent_off   = (int*)(s_scores + B_SENT);         // 1280 ints
  float* partial    = (float*)(sent_off + B_SENT);       // 655360

  // 1) both GEMMs with f32 WMMA (exact f32): one wave per 16x16 tile
  gemm_bias_wmma<<<128, 32, 0, stream>>>(s_t_hat, Wd, bd, sent_s_t_hat, Wsd, bsd,
                                         dec_fea, sdec_fea);

  // 2) word-level scores: 81920 rows, one wave each (doc = row / 2560)
  add_attn_scores<<<(B_SENT * T_K) / 8, 256, 0, stream>>>(enc_feature, dec_fea, v_w,
                                                          scores, B_SENT * T_K, S * T_K);
  // 3) sentence-level scores: 1280 rows (doc = row / 40)
  add_attn_scores<<<B_SENT / 8, 256, 0, stream>>>(sent_enc_feat, sdec_fea, sv_w,
                                                  s_scores, B_SENT, S);

  // 4) softmaxes
  word_softmax<<<B_SENT / 8, 256, 0, stream>>>(scores, seq_lens2, w_attn);
  sent_softmax<<<B_DOC / 8, 256, 0, stream>>>(s_scores, sent_pad_mask, out_sattn);

  // 5) per-doc word offsets
  doc_offsets<<<1, 32, 0, stream>>>(seq_lens2, sent_off);

  // 6) attn_dist_sw: zero then scatter valid words
  fill_zero<<<(B_DOC * MAXDOC + 255) / 256, 256, 0, stream>>>(out_attn, B_DOC * MAXDOC);
  scatter_attn<<<(B_SENT * T_K + 255) / 256, 256, 0, stream>>>(w_attn, out_sattn,
                                                               seq_lens2, sent_off, out_attn);

  // 7) c_t: per-sentence weighted sums, then 40-way reduce per doc
  ct_partial<<<B_SENT, 256, 0, stream>>>(enc_outputs, w_attn, out_sattn, seq_lens2, partial);
  ct_reduce<<<(B_DOC * N + 255) / 256, 256, 0, stream>>>(partial, out_ct);

  // 8) coverage passthrough
  hipMemcpyAsync(out_cov, coverage, (size_t)B_SENT * T_K * sizeof(float),
                 hipMemcpyDeviceToDevice, stream);
}